// DistinaNet_54752243089626
// MI455X (gfx1250) — compile-verified
//
#include <hip/hip_runtime.h>
#include <hip/hip_bf16.h>
#include <stdint.h>
#include <math.h>

#define NUM_CLASSES 80
#define PRE_K 1000
#define MAX_DET 100
#define CHUNK 8192
#define NTHREADS 1024
#define TR_ROWS 256
#define SCORE_THR 0.05f
#define IOU_THR 0.5f

typedef unsigned int uint;
typedef unsigned long long u64;
typedef unsigned int v4u __attribute__((ext_vector_type(4)));
typedef int v8i __attribute__((ext_vector_type(8)));
typedef int v4i __attribute__((ext_vector_type(4)));

// ---------------------------------------------------------------------------
// TDM load into LDS. Two shapes:
//   stride_elems == 1 : one contiguous run of `tile_rows` floats
//                       (tile_dim0 = tile_rows, tile_dim1 = 1)
//   stride_elems >  1 : column gather, one 4B element per stride-sized row
//                       (tile_dim0 = 1, tile_dim1 = tile_rows)
// Elements beyond `rows_remaining` read as zero per D# OOB rules (harmless:
// 0.0f < SCORE_THR).  D# layout per CDNA5 ISA ch.8.
// ---------------------------------------------------------------------------
__device__ __forceinline__ void tdm_load_col(const float* gaddr, uint lds_off,
                                             uint rows_remaining, uint tile_rows,
                                             uint stride_elems)
{
    u64 ga = (u64)(uintptr_t)gaddr;
    v4u g0;
    g0.x = 1u;                                   // count=1 (valid), user mode
    g0.y = lds_off;                              // lds_addr (bytes)
    g0.z = (uint)ga;                             // global_addr[31:0]
    g0.w = (uint)((ga >> 32) & 0x01FFFFFFu)      // global_addr[56:32]
           | 0x80000000u;                        // type=2 in bits[127:126]

    uint d0, d1, t0, t1;
    u64  s0;
    if (stride_elems == 1u) {                    // contiguous burst
        d0 = rows_remaining; d1 = 1u;
        t0 = tile_rows;      t1 = 1u;
        s0 = 0;
    } else {                                     // strided column gather
        d0 = 1u;             d1 = rows_remaining;
        t0 = 1u;             t1 = tile_rows;
        s0 = (u64)stride_elems;
    }
    const uint t2 = 0u;

    v8i g1;
    g1[0] = (int)0x00020000u;                                   // data_size=2 (4B)
    g1[1] = (int)((d0 & 0xFFFFu) << 16);
    g1[2] = (int)(((d0 >> 16) & 0xFFFFu) | ((d1 & 0xFFFFu) << 16));
    g1[3] = (int)(((d1 >> 16) & 0xFFFFu) | ((t0 & 0xFFFFu) << 16));
    g1[4] = (int)((t1 & 0xFFFFu) | ((t2 & 0xFFFFu) << 16));
    g1[5] = (int)(uint)(s0 & 0xFFFFFFFFu);
    g1[6] = (int)((uint)(s0 >> 32) & 0xFFFFu);
    g1[7] = 0;

    v4i g2 = {0, 0, 0, 0};
    v4i g3 = {0, 0, 0, 0};
#if defined(__clang_major__) && __clang_major__ >= 23
    v8i g4 = {0, 0, 0, 0, 0, 0, 0, 0};
    __builtin_amdgcn_tensor_load_to_lds(g0, g1, g2, g3, g4, 0);
#else
    __builtin_amdgcn_tensor_load_to_lds(g0, g1, g2, g3, 0);
#endif
}

// ---------------------------------------------------------------------------
// Transpose classification (A,80) -> (80,A) so per-class scans are contiguous.
// LDS tile padded to 81 columns -> conflict-free column reads (gcd(81,64)=1).
// ---------------------------------------------------------------------------
__global__ __launch_bounds__(TR_ROWS)
void transpose_cls_kernel(const float* __restrict__ src,
                          float* __restrict__ dstT, int A)
{
    __shared__ float tile[TR_ROWS * 81];
    const int t  = (int)threadIdx.x;
    const int a0 = (int)blockIdx.x * TR_ROWS;
    const long long lim = (long long)(A - a0) * NUM_CLASSES;   // valid elems in tile

    // coalesced load of TR_ROWS*80 contiguous floats
    for (int k = 0; k < NUM_CLASSES; ++k) {
        int i = k * TR_ROWS + t;                 // [0, 80*256)
        int r = i / NUM_CLASSES, c2 = i % NUM_CLASSES;
        float v = ((long long)i < lim) ? src[(long long)a0 * NUM_CLASSES + i] : 0.0f;
        tile[r * 81 + c2] = v;
    }
    __syncthreads();
    // coalesced store of 80 columns of TR_ROWS floats
    for (int k = 0; k < NUM_CLASSES; ++k) {
        int o  = k * TR_ROWS + t;
        int c2 = o >> 8, r = o & (TR_ROWS - 1);
        if (a0 + r < A)
            dstT[(long long)c2 * A + a0 + r] = tile[r * 81 + c2];
    }
}

__global__ __launch_bounds__(NTHREADS)
void detinanet_postproc_kernel(const float* __restrict__ colsrc, int colstride,
                               const float* __restrict__ reg,
                               const float* __restrict__ dst,
                               const float* __restrict__ anc,
                               const int* __restrict__ imh_p,
                               const int* __restrict__ imw_p,
                               float* __restrict__ out,
                               int A)
{
    __shared__ float stage[2][CHUNK];            // 64 KB double-buffered TDM staging
    __shared__ uint  hist[256];
    __shared__ u64   keys[NTHREADS];             // (score_bits<<32)|~index
    __shared__ uint  elist[64];
    __shared__ float bx1[NTHREADS], by1[NTHREADS], bx2[NTHREADS], by2[NTHREADS];
    __shared__ float barea[NTHREADS], bscore[NTHREADS], bdist[NTHREADS];
    __shared__ uint  keep[NTHREADS];
    __shared__ uint  outpos[MAX_DET];
    __shared__ uint  sh[8];  // 0:gcount 1:ecount 2:prefix 3:kth 4:shortfall 5:N 6:outcnt

    const int  tid = (int)threadIdx.x;
    const int  c   = (int)blockIdx.x;
    const bool w0  = (tid < 32);                 // wave 0 issues DMA
    const float img_w = (float)imw_p[0];
    const float img_h = (float)imh_p[0];
    const uint  thrbits = __float_as_uint(SCORE_THR);
    const float* colbase = (colstride == 1) ? (colsrc + (u64)c * (u64)A)
                                            : (colsrc + c);
    const int  nchunk = (A + CHUNK - 1) / CHUNK;
    const uint lds0 = (uint)(uintptr_t)&stage[0][0];
    const uint lds1 = (uint)(uintptr_t)&stage[1][0];

    // Stream the class column through LDS via TDM, double buffered.
    auto scan = [&](auto body) {
        if (w0)
            tdm_load_col(colbase, lds0, (uint)A, CHUNK, (uint)colstride);
        for (int ck = 0; ck < nchunk; ++ck) {
            if (w0) {
                if (ck + 1 < nchunk) {
                    tdm_load_col(colbase + (u64)(ck + 1) * CHUNK * (u64)colstride,
                                 ((ck + 1) & 1) ? lds1 : lds0,
                                 (uint)(A - (ck + 1) * CHUNK), CHUNK, (uint)colstride);
                    __builtin_amdgcn_s_wait_tensorcnt(1);   // chunk ck done
                } else {
                    __builtin_amdgcn_s_wait_tensorcnt(0);
                }
            }
            __syncthreads();
            body((const float*)stage[ck & 1], ck * CHUNK);
            __syncthreads();
        }
    };

    // -------- Phase 1: exact top-1000 threshold via 4-pass MSB radix select
    if (tid == 0) { sh[2] = 0u; sh[3] = PRE_K; sh[4] = 0u; }
    __syncthreads();

    for (int p = 0; p < 4; ++p) {
        const uint shift  = 24u - 8u * (uint)p;
        const uint prefix = sh[2];
        const bool sf     = (sh[4] != 0u);
        const uint himask = (p == 0) ? 0u : (0xFFFFFFFFu << (shift + 8u));
        if (tid < 256) hist[tid] = 0u;
        __syncthreads();
        if (!sf) {
            scan([&](const float* buf, int rowbase) {
                for (int e = tid; e < CHUNK; e += NTHREADS) {
                    if (rowbase + e < A) {
                        float v = buf[e];
                        if (v > SCORE_THR) {
                            uint b = __float_as_uint(v);
                            if ((b & himask) == (prefix & himask))
                                atomicAdd(&hist[(b >> shift) & 0xFFu], 1u);
                        }
                    }
                }
            });
            if (tid == 0) {
                uint kth = sh[3], cum = 0u;
                bool found = false;
                for (int b = 255; b >= 0; --b) {
                    uint nc = cum + hist[b];
                    if (nc >= kth) {            // pivot byte found
                        sh[2] = prefix | ((uint)b << shift);
                        sh[3] = kth - cum;      // rank among == pivot prefix
                        found = true;
                        break;
                    }
                    cum = nc;
                }
                if (!found) sh[4] = 1u;         // < 1000 pass the score threshold
            }
        }
        __syncthreads();
    }

    const uint Tbits = (sh[4] != 0u) ? thrbits : sh[2];

    // -------- Phase 2: collect candidates
    if (tid == 0) { sh[0] = 0u; sh[1] = 0u; }
    __syncthreads();
    scan([&](const float* buf, int rowbase) {
        for (int e = tid; e < CHUNK; e += NTHREADS) {
            int row = rowbase + e;
            if (row < A) {
                float v = buf[e];
                if (v > SCORE_THR) {
                    uint b = __float_as_uint(v);
                    if (b > Tbits) {
                        uint pos = atomicAdd(&sh[0], 1u);
                        if (pos < (uint)NTHREADS)
                            keys[pos] = ((u64)b << 32) | (u64)(uint)(~(uint)row);
                    } else if (b == Tbits) {
                        uint pos = atomicAdd(&sh[1], 1u);
                        if (pos < 64u) elist[pos] = (uint)row;
                    }
                }
            }
        }
    });
    if (tid == 0) {
        uint g  = sh[0]; if (g > (uint)PRE_K) g = (uint)PRE_K;
        uint ne = sh[1]; if (ne > 64u) ne = 64u;
        // ties at pivot: ascending index, as lax.top_k orders equals
        for (int i = 1; i < (int)ne; ++i) {
            uint v = elist[i]; int j = i - 1;
            while (j >= 0 && elist[j] > v) { elist[j + 1] = elist[j]; --j; }
            elist[j + 1] = v;
        }
        uint need = (g < (uint)PRE_K) ? ((uint)PRE_K - g) : 0u;
        if (need > ne) need = ne;
        for (uint i = 0; i < need; ++i)
            keys[g + i] = ((u64)Tbits << 32) | (u64)(uint)(~elist[i]);
        sh[5] = g + need;
    }
    __syncthreads();
    const uint N = sh[5];
    if ((uint)tid >= N) keys[tid] = 0ull;
    __syncthreads();

    // -------- Phase 3: bitonic sort 1024 keys, descending
    for (uint k = 2; k <= (uint)NTHREADS; k <<= 1) {
        for (uint j = k >> 1; j > 0; j >>= 1) {
            uint i = (uint)tid, ixj = i ^ j;
            if (ixj > i) {
                u64 a = keys[i], b = keys[ixj];
                bool sw = ((i & k) == 0u) ? (a < b) : (a > b);
                if (sw) { keys[i] = b; keys[ixj] = a; }
            }
            __syncthreads();
        }
    }

    // -------- Phase 4: gather + decode the survivors
    {
        u64  key = keys[tid];
        bool valid = ((uint)tid < N);
        keep[tid] = valid ? 1u : 0u;
        if (valid) {
            uint idx = ~(uint)key;
            float a0 = anc[idx * 4 + 0], a1 = anc[idx * 4 + 1];
            float a2 = anc[idx * 4 + 2], a3 = anc[idx * 4 + 3];
            float r0 = reg[idx * 4 + 0], r1 = reg[idx * 4 + 1];
            float r2 = reg[idx * 4 + 2], r3 = reg[idx * 4 + 3];
            float w  = a2 - a0, h = a3 - a1;
            float cx = a0 + 0.5f * w, cy = a1 + 0.5f * h;
            float pcx = cx + (r0 * 0.1f) * w;
            float pcy = cy + (r1 * 0.1f) * h;
            float pw  = expf(r2 * 0.2f) * w;
            float ph  = expf(r3 * 0.2f) * h;
            float x1 = fminf(fmaxf(pcx - 0.5f * pw, 0.0f), img_w);
            float y1 = fminf(fmaxf(pcy - 0.5f * ph, 0.0f), img_h);
            float x2 = fminf(fmaxf(pcx + 0.5f * pw, 0.0f), img_w);
            float y2 = fminf(fmaxf(pcy + 0.5f * ph, 0.0f), img_h);
            bx1[tid] = x1; by1[tid] = y1; bx2[tid] = x2; by2[tid] = y2;
            barea[tid]  = (x2 - x1) * (y2 - y1);
            bscore[tid] = __uint_as_float((uint)(key >> 32));
            bdist[tid]  = dst[idx];
        } else {
            bx1[tid] = 0.f; by1[tid] = 0.f; bx2[tid] = 0.f; by2[tid] = 0.f;
            barea[tid] = 0.f; bscore[tid] = 0.f; bdist[tid] = 0.f;
        }
    }
    __syncthreads();

    // -------- Phase 5: greedy NMS.  A suppressed pivot writes nothing, so the
    // barrier is only needed after active pivots (keep[i] is uniform: last
    // write to keep[] was before the previous barrier).
    int Nn = (int)N; if (Nn > PRE_K) Nn = PRE_K;
    for (int i = 0; i < Nn; ++i) {
        if (keep[i]) {
            if (tid > i && tid < Nn && keep[tid]) {
                float xx1 = fmaxf(bx1[i], bx1[tid]);
                float yy1 = fmaxf(by1[i], by1[tid]);
                float xx2 = fminf(bx2[i], bx2[tid]);
                float yy2 = fminf(by2[i], by2[tid]);
                float iw = fmaxf(xx2 - xx1, 0.0f);
                float ih = fmaxf(yy2 - yy1, 0.0f);
                float inter = iw * ih;
                float iou = inter / (barea[i] + barea[tid] - inter + 1e-8f);
                if (iou > IOU_THR) keep[tid] = 0u;
            }
            __syncthreads();
        }
    }
    __syncthreads();

    // -------- Phase 6: compact kept (already score-sorted) and emit top-100
    if (tid == 0) {
        uint cnt = 0u;
        for (int t = 0; t < Nn && cnt < (uint)MAX_DET; ++t)
            if (keep[t]) outpos[cnt++] = (uint)t;
        sh[6] = cnt;
    }
    __syncthreads();
    if (tid < MAX_DET) {
        const uint cnt = sh[6];
        const int  o   = c * MAX_DET + tid;
        float s = 0.f, x1 = 0.f, y1 = 0.f, x2 = 0.f, y2 = 0.f, dd = 0.f;
        int ci = -1;
        if ((uint)tid < cnt) {
            uint t = outpos[tid];
            s = bscore[t]; x1 = bx1[t]; y1 = by1[t]; x2 = bx2[t]; y2 = by2[t];
            dd = bdist[t]; ci = c;
        }
        out[o] = s;                                              // scores    [0, 8000)
        ((int*)out)[NUM_CLASSES * MAX_DET + o] = ci;             // class ids [8000, 16000)
        float* ob = out + 2 * NUM_CLASSES * MAX_DET;             // boxes     [16000, 48000)
        ob[o * 4 + 0] = x1; ob[o * 4 + 1] = y1;
        ob[o * 4 + 2] = x2; ob[o * 4 + 3] = y2;
        out[6 * NUM_CLASSES * MAX_DET + o] = dd;                 // distances [48000, 56000)
    }
}

extern "C" void kernel_launch(void* const* d_in, const int* in_sizes, int n_in,
                              void* d_out, int out_size, void* d_ws, size_t ws_size,
                              hipStream_t stream) {
    const float* cls = (const float*)d_in[0];
    const float* reg = (const float*)d_in[1];
    const float* dst = (const float*)d_in[2];
    const float* anc = (const float*)d_in[3];
    const int*   imh = (const int*)d_in[4];
    const int*   imw = (const int*)d_in[5];
    const int A = in_sizes[1] / 4;   // regression is (1, A, 4)

    const size_t needed = (size_t)A * NUM_CLASSES * sizeof(float);
    const float* colsrc = cls;
    int colstride = NUM_CLASSES;
    if (d_ws != nullptr && ws_size >= needed) {
        // Transpose once: turns 80x5 strided 4B-gather scans into contiguous
        // 32KB TDM bursts.  Path choice depends only on launch params.
        transpose_cls_kernel<<<(A + TR_ROWS - 1) / TR_ROWS, TR_ROWS, 0, stream>>>(
            cls, (float*)d_ws, A);
        colsrc = (const float*)d_ws;
        colstride = 1;
    }
    detinanet_postproc_kernel<<<NUM_CLASSES, NTHREADS, 0, stream>>>(
        colsrc, colstride, reg, dst, anc, imh, imw, (float*)d_out, A);
}